// BahdanauAttention_50465865728414
// MI455X (gfx1250) — compile-verified
//
#include <hip/hip_runtime.h>
#include <hip/hip_bf16.h>

#define B_   32
#define A_   512
#define EH_  256
#define DH_  384
#define HW_  4096   // 64*64

typedef __attribute__((ext_vector_type(2))) float v2f;
typedef __attribute__((ext_vector_type(8))) float v8f;

// CDNA5 hardware tanh (gfx1250 V_TANH_F32). TRANS op: ISA requires 1
// independent op or V_NOP after a trans op before its output is consumed
// (hazard not hardware-interlocked for co-executing multicycle ops), so we
// bundle a v_nop into the asm.
__device__ __forceinline__ float hw_tanhf(float x) {
    float r;
    asm volatile("v_tanh_f32 %0, %1\n\tv_nop" : "=v"(r) : "v"(x));
    return r;
}

// ---------------------------------------------------------------------------
// Kernel 1: dec_proj[B_, A_] = dec_h[B_, DH_] @ W_dec[A_, DH_]^T + b_dec
// One wave computes one 16x16 tile via V_WMMA_F32_16X16X4_F32 (f32 in/out ==
// reference precision). Grid = (B_/16) * (A_/16) = 2*32 = 64 waves.
// ---------------------------------------------------------------------------
__global__ __launch_bounds__(32) void dec_proj_wmma_kernel(
        const float* __restrict__ dec_h,
        const float* __restrict__ W_dec,
        const float* __restrict__ b_dec,
        float* __restrict__ dec_proj) {
    const int tile = blockIdx.x;
    const int m0   = (tile & 1) * 16;    // row tile in B_
    const int n0   = (tile >> 1) * 16;   // col tile in A_
    const int lane = threadIdx.x;
    const int half = lane >> 4;          // selects K pair {0,1} vs {2,3}
    const int l15  = lane & 15;

    // A-frag: lane l(<16) holds A[M=l, K=k..k+1]; lane l+16 holds K=k+2..k+3
    const float* arow = dec_h + (m0 + l15) * DH_;
    // B-frag (B = W_dec^T, K x N): lane holds N=l15, same K-half split
    const float* brow = W_dec + (n0 + l15) * DH_;

    v8f c = {};
    for (int k = 0; k < DH_; k += 4) {
        const int kb = k + 2 * half;
        v2f a; a.x = arow[kb]; a.y = arow[kb + 1];
        v2f b; b.x = brow[kb]; b.y = brow[kb + 1];
        c = __builtin_amdgcn_wmma_f32_16x16x4_f32(
                /*neg_a=*/false, a, /*neg_b=*/false, b,
                /*c_mod=*/(short)0, c, /*reuse_a=*/false, /*reuse_b=*/false);
    }

    // C/D layout: lanes 0-15 -> M = v, N = lane; lanes 16-31 -> M = v+8
    const float bias = b_dec[n0 + l15];
    #pragma unroll
    for (int v = 0; v < 8; ++v) {
        const int row = m0 + v + 8 * half;
        const int col = n0 + l15;
        dec_proj[row * A_ + col] = c[v] + bias;
    }
}

// ---------------------------------------------------------------------------
// Kernel 2 (bandwidth-dominant, 268 MB stream):
// scores[b,hw] = b_energy + sum_a w_energy[a]*tanh(weo[b,a,hw] + dec_proj[b,a])
// Grid = B_*8 blocks of 128 threads; each thread owns 4 contiguous pixels
// (float4 / global_load_b128), loops over all 512 channels. Block-uniform
// dec_proj row + w_energy staged in LDS once.
// ---------------------------------------------------------------------------
__global__ __launch_bounds__(128) void scores_kernel(
        const float* __restrict__ weo,
        const float* __restrict__ dec_proj,
        const float* __restrict__ w_energy,
        const float* __restrict__ b_energy,
        float* __restrict__ scores) {
    __shared__ float s_dp[A_];
    __shared__ float s_we[A_];

    const int b     = blockIdx.x >> 3;   // 8 chunks of 512 px per batch
    const int chunk = blockIdx.x & 7;
    const int t     = threadIdx.x;

    for (int i = t; i < A_; i += 128) {
        s_dp[i] = dec_proj[b * A_ + i];
        s_we[i] = w_energy[i];
    }
    __syncthreads();

    const int px = chunk * 512 + t * 4;
    const float4* base =
        (const float4*)(weo + (size_t)b * A_ * HW_ + px);

    float acc0 = 0.f, acc1 = 0.f, acc2 = 0.f, acc3 = 0.f;
    #pragma unroll 4
    for (int a = 0; a < A_; ++a) {
        const float4 v  = base[a * (HW_ / 4)];
        const float  dp = s_dp[a];
        const float  we = s_we[a];
        acc0 += we * hw_tanhf(v.x + dp);
        acc1 += we * hw_tanhf(v.y + dp);
        acc2 += we * hw_tanhf(v.z + dp);
        acc3 += we * hw_tanhf(v.w + dp);
    }

    const float be = b_energy[0];
    float4 out; out.x = acc0 + be; out.y = acc1 + be;
    out.z = acc2 + be; out.w = acc3 + be;
    *(float4*)(scores + b * HW_ + px) = out;
}

// ---------------------------------------------------------------------------
// Kernel 3: softmax over the 4096 spatial scores of each batch.
// One 256-thread block (8 waves) per batch; 16 values/thread kept in regs,
// wave shuffle + LDS tree for max and sum.
// ---------------------------------------------------------------------------
__global__ __launch_bounds__(256) void softmax_kernel(
        const float* __restrict__ scores,
        float* __restrict__ alphas) {
    __shared__ float red[8];
    const int b = blockIdx.x;
    const int t = threadIdx.x;
    const float* s = scores + b * HW_;

    float v[16];
    float m = -INFINITY;
    #pragma unroll
    for (int i = 0; i < 16; ++i) { v[i] = s[t + i * 256]; m = fmaxf(m, v[i]); }
    #pragma unroll
    for (int off = 16; off; off >>= 1) m = fmaxf(m, __shfl_xor(m, off, 32));
    if ((t & 31) == 0) red[t >> 5] = m;
    __syncthreads();
    float bm = red[0];
    #pragma unroll
    for (int w = 1; w < 8; ++w) bm = fmaxf(bm, red[w]);

    float sum = 0.f;
    #pragma unroll
    for (int i = 0; i < 16; ++i) { v[i] = __expf(v[i] - bm); sum += v[i]; }
    #pragma unroll
    for (int off = 16; off; off >>= 1) sum += __shfl_xor(sum, off, 32);
    __syncthreads();
    if ((t & 31) == 0) red[t >> 5] = sum;
    __syncthreads();
    float bs = 0.f;
    #pragma unroll
    for (int w = 0; w < 8; ++w) bs += red[w];

    const float inv = 1.0f / bs;
    #pragma unroll
    for (int i = 0; i < 16; ++i) alphas[b * HW_ + t + i * 256] = v[i] * inv;
}

// ---------------------------------------------------------------------------
// Kernel 4 (134 MB stream): context[b,c] = sum_hw enc_out[b,c,hw]*alphas[b,hw]
// Block = 256 threads = 8 waves; block handles 8 channels of one batch with
// alphas[b,:] staged in LDS (16 KB). Each wave reduces one channel with
// float4 loads + wave32 shuffle reduction.
// ---------------------------------------------------------------------------
__global__ __launch_bounds__(256) void context_kernel(
        const float* __restrict__ enc_out,
        const float* __restrict__ alphas,
        float* __restrict__ context) {
    __shared__ float s_al[HW_];
    const int b  = blockIdx.x >> 5;          // EH_/8 = 32 blocks per batch
    const int c0 = (blockIdx.x & 31) * 8;
    const int t  = threadIdx.x;

    for (int i = t; i < HW_; i += 256) s_al[i] = alphas[b * HW_ + i];
    __syncthreads();

    const int wave = t >> 5, lane = t & 31;
    const int c = c0 + wave;
    const float4* e =
        (const float4*)(enc_out + ((size_t)b * EH_ + c) * HW_);
    const float4* al = (const float4*)s_al;

    float acc = 0.f;
    #pragma unroll 4
    for (int i = 0; i < HW_ / 4; i += 32) {
        const float4 ev = e[i + lane];
        const float4 av = al[i + lane];
        acc += ev.x * av.x + ev.y * av.y + ev.z * av.z + ev.w * av.w;
    }
    #pragma unroll
    for (int off = 16; off; off >>= 1) acc += __shfl_xor(acc, off, 32);
    if (lane == 0) context[b * EH_ + c] = acc;
}

// ---------------------------------------------------------------------------
extern "C" void kernel_launch(void* const* d_in, const int* in_sizes, int n_in,
                              void* d_out, int out_size, void* d_ws, size_t ws_size,
                              hipStream_t stream) {
    const float* dec_h  = (const float*)d_in[0];   // [B, DH]
    const float* weo    = (const float*)d_in[1];   // [B, A, H, W]
    const float* enc    = (const float*)d_in[2];   // [B, EH, H, W]
    const float* W_dec  = (const float*)d_in[3];   // [A, DH]
    const float* b_dec  = (const float*)d_in[4];   // [A]
    const float* w_en   = (const float*)d_in[5];   // [A]
    const float* b_en   = (const float*)d_in[6];   // scalar

    float* out      = (float*)d_out;
    float* context  = out;                 // [B, EH]
    float* alphas   = out + B_ * EH_;      // [B, H, W]

    float* dec_proj = (float*)d_ws;        // [B, A]   = 64 KB
    float* scores   = dec_proj + B_ * A_;  // [B, HW]  = 512 KB

    dec_proj_wmma_kernel<<<(B_ / 16) * (A_ / 16), 32, 0, stream>>>(
        dec_h, W_dec, b_dec, dec_proj);
    scores_kernel<<<B_ * 8, 128, 0, stream>>>(weo, dec_proj, w_en, b_en, scores);
    softmax_kernel<<<B_, 256, 0, stream>>>(scores, alphas);
    context_kernel<<<B_ * (EH_ / 8), 256, 0, stream>>>(enc, alphas, context);
}